// Patch_Drop_66606352827266
// MI455X (gfx1250) — compile-verified
//
#include <hip/hip_runtime.h>

// ---------------------------------------------------------------------------
// Patch-Drop for MI455X (gfx1250, wave32).
// b=8, c=64, H=W=512, ps=32 -> L=256 patches, ps2=1024, k=103.
//
// Key algebraic reduction: the reference's scrambled einsum weight is
// g[b, (row&1)*32 + col], so per patch we only need 64 partial sums s[b,l,q]
// (q = parity*32+col, summed over channels), the patch sum-of-squares ss,
// and the per-channel means g. num = S(256x64) . g is done with
// V_WMMA_F32_16X16X4_F32 (f32 keeps top-k ranking exact).
// All reductions are fixed-order (no float atomics) => deterministic output.
// ---------------------------------------------------------------------------

typedef __attribute__((ext_vector_type(2))) float v2f;
typedef __attribute__((ext_vector_type(8))) float v8f;

#define NB    8
#define NC    64
#define NH    512
#define NW    512
#define PS    32
#define PS2   1024
#define NL    256          // (512/32)^2
#define TOPK  103          // int(256*0.4)+1
#define INV_HW (1.0f / 262144.0f)
#define DROPP 0.15f
#define KEEPS (1.0f / 0.85f)

// ---------------------------------------------------------------------------
// Kernel 1: per-patch stats. One block per (b,l): reads the 64ch x 32x32 tile
// (256 KB) with float4 loads, produces:
//   psum[b][l][ch] : per-channel patch sum (for g, reduced later)
//   s[b][l][q]     : q = (row&1)*32+col, summed over channels
//   ss[b][l]       : sum of squares over whole tile
// ---------------------------------------------------------------------------
__global__ __launch_bounds__(256) void k_stats(const float* __restrict__ x,
                                               float* __restrict__ psum,
                                               float* __restrict__ s,
                                               float* __restrict__ ss) {
  const int blk = blockIdx.x;          // b*NL + l
  const int b = blk >> 8, l = blk & 255;
  const int hn = l >> 4, wn = l & 15;
  const int wave = threadIdx.x >> 5, lane = threadIdx.x & 31;
  const int rowoff = lane >> 3;        // 0..3 (row parity fixed per lane)
  const int quad = lane & 7;           // 0..7 -> columns quad*4..quad*4+3

  const float* base = x + (size_t)b * (NC * NH * NW)
                        + (size_t)(hn * PS) * NW + wn * PS;

  float4 acc = make_float4(0.f, 0.f, 0.f, 0.f);   // per-column partial sums
  float4 acc2 = make_float4(0.f, 0.f, 0.f, 0.f);  // squares

  __shared__ float sbuf[8][32][4];
  __shared__ float ssbuf[8][32];

  for (int cc = 0; cc < 8; ++cc) {
    const int ch = wave * 8 + cc;
    const float* cb = base + (size_t)ch * (NH * NW);
    float chacc = 0.f;
#pragma unroll
    for (int rb = 0; rb < 8; ++rb) {
      const int r = rb * 4 + rowoff;
      const float4 v = *(const float4*)(cb + r * NW + quad * 4);
      acc.x += v.x; acc.y += v.y; acc.z += v.z; acc.w += v.w;
      acc2.x += v.x * v.x; acc2.y += v.y * v.y;
      acc2.z += v.z * v.z; acc2.w += v.w * v.w;
      chacc += (v.x + v.y) + (v.z + v.w);
    }
    // deterministic butterfly reduce across the wave -> full 32x32 channel sum
#pragma unroll
    for (int off = 16; off; off >>= 1) chacc += __shfl_xor(chacc, off, 32);
    if (lane == 0) psum[(size_t)blk * NC + ch] = chacc;
  }

  sbuf[wave][lane][0] = acc.x; sbuf[wave][lane][1] = acc.y;
  sbuf[wave][lane][2] = acc.z; sbuf[wave][lane][3] = acc.w;
  ssbuf[wave][lane] = (acc2.x + acc2.y) + (acc2.z + acc2.w);
  __syncthreads();

  const int t = threadIdx.x;
  if (t < 64) {
    // q = t: parity p = t>>5, column col = t&31
    const int p = t >> 5, col = t & 31;
    const int q2 = col >> 2, j = col & 3;
    const int l1 = (p << 3) | q2;          // lanes with this parity/quad
    const int l2 = ((p + 2) << 3) | q2;
    float a = 0.f;
#pragma unroll
    for (int w = 0; w < 8; ++w) a += sbuf[w][l1][j] + sbuf[w][l2][j];
    s[(size_t)blk * 64 + t] = a;
  }
  if (t == 0) {
    float a = 0.f;
    for (int w = 0; w < 8; ++w)
      for (int i = 0; i < 32; ++i) a += ssbuf[w][i];
    ss[blk] = a;
  }
}

// ---------------------------------------------------------------------------
// Kernel 2: per batch (8 blocks): g from psum (fixed-order), then
// num = S(256x64) . g via V_WMMA_F32_16X16X4_F32 (g replicated across the 16
// B-columns so every D column equals num), then sim + exact rank top-k.
// ---------------------------------------------------------------------------
__global__ __launch_bounds__(256) void k_select(const float* __restrict__ psum,
                                                const float* __restrict__ s,
                                                const float* __restrict__ ss,
                                                int* __restrict__ sel) {
  const int b = blockIdx.x;
  const int t = threadIdx.x;
  const int lane = t & 31;
  const int wave = t >> 5;

  __shared__ float gL[64];
  __shared__ float numL[NL];
  __shared__ float simL[NL];
  __shared__ float ngSh;

  if (t < 64) {
    float a = 0.f;
    const float* p = psum + (size_t)b * NL * NC + t;
    for (int l = 0; l < NL; ++l) a += p[l * NC];   // deterministic order
    gL[t] = a * INV_HW;
  }
  __syncthreads();
  if (t < 32) {
    float gg = gL[t] * gL[t] + gL[t + 32] * gL[t + 32];
#pragma unroll
    for (int off = 16; off; off >>= 1) gg += __shfl_xor(gg, off, 32);
    if (t == 0) ngSh = 32.0f * sqrtf(gg);          // ||g_expanded|| = ps*||g||
  }
  __syncthreads();

  // --- WMMA matvec: each wave computes 2 tiles of 16 rows of num ---
  const float* Sb = s + (size_t)b * NL * 64;
  const int m = lane & 15;                 // A: M = lane%16
  const int koff = (lane >> 4) << 1;       // A/B: K offset 0 or 2 per lane half
  for (int tile = wave * 2; tile < wave * 2 + 2; ++tile) {
    const int row0 = tile * 16;
    v8f cacc = {};
#pragma unroll
    for (int kc = 0; kc < 16; ++kc) {
      const int kb = kc * 4 + koff;
      v2f av, bv;
      av.x = Sb[(row0 + m) * 64 + kb];
      av.y = Sb[(row0 + m) * 64 + kb + 1];
      bv.x = gL[kb];                       // B[k][n] = g[k] for all n
      bv.y = gL[kb + 1];
      cacc = __builtin_amdgcn_wmma_f32_16x16x4_f32(
          false, av, false, bv, (short)0, cacc, false, false);
    }
    if (m == 0) {                          // lanes with N==0: lane 0 (M=j), lane 16 (M=8+j)
      const int mb = row0 + ((lane >> 4) << 3);
#pragma unroll
      for (int j = 0; j < 8; ++j) numL[mb + j] = cacc[j];
    }
  }
  __syncthreads();

  const float np = sqrtf(ss[b * NL + t]);
  const float den = fmaxf(ngSh * np, 1e-8f);
  simL[t] = numL[t] / den;
  __syncthreads();

  // exact jax.lax.top_k semantics: descending, ties -> lower index first
  const float my = simL[t];
  int cnt = 0;
  for (int j = 0; j < NL; ++j) {
    const float sj = simL[j];
    cnt += (sj > my) || (sj == my && j < t);
  }
  sel[b * NL + t] = (cnt < TOPK) ? 1 : 0;
}

// ---------------------------------------------------------------------------
// Kernel 3: apply. One block per (b,l,ch) 32x32 plane; float4 in/out.
// Unselected patches (uniform per-block branch) never read noise.
// ---------------------------------------------------------------------------
__global__ __launch_bounds__(256) void k_apply(const float* __restrict__ x,
                                               const float* __restrict__ noise,
                                               const int* __restrict__ sel,
                                               float* __restrict__ out) {
  const int gid = blockIdx.x;              // b*NL*NC + l*NC + ch
  const int ch = gid & 63;
  const int l = (gid >> 6) & 255;
  const int b = gid >> 14;
  const int hn = l >> 4, wn = l & 15;
  const int t = threadIdx.x;
  const int r = t >> 3, quad = t & 7;

  const size_t xoff = (size_t)b * (NC * NH * NW) + (size_t)ch * (NH * NW)
                    + (size_t)(hn * PS + r) * NW + wn * PS + quad * 4;
  const float4 v = *(const float4*)(x + xoff);
  float4 o;
  if (sel[b * NL + l]) {
    const size_t noff = ((size_t)(b * NL + l) * NC + ch) * PS2 + r * PS + quad * 4;
    const float4 n = *(const float4*)(noise + noff);
    o.x = v.x * ((n.x >= DROPP) ? KEEPS : 0.0f);
    o.y = v.y * ((n.y >= DROPP) ? KEEPS : 0.0f);
    o.z = v.z * ((n.z >= DROPP) ? KEEPS : 0.0f);
    o.w = v.w * ((n.w >= DROPP) ? KEEPS : 0.0f);
  } else {
    o = v;
  }
  *(float4*)(out + xoff) = o;
}

// ---------------------------------------------------------------------------
extern "C" void kernel_launch(void* const* d_in, const int* in_sizes, int n_in,
                              void* d_out, int out_size, void* d_ws, size_t ws_size,
                              hipStream_t stream) {
  const float* x = (const float*)d_in[0];
  const float* noise = (const float*)d_in[1];
  (void)in_sizes; (void)n_in; (void)out_size; (void)ws_size;

  float* ws = (float*)d_ws;
  float* psum = ws;                        // NB*NL*NC = 131072 floats
  float* s    = ws + 131072;               // NB*NL*64 = 131072 floats
  float* ss   = ws + 262144;               // NB*NL    =   2048 floats
  int*   sel  = (int*)(ws + 264192);       // NB*NL    =   2048 ints

  k_stats<<<NB * NL, 256, 0, stream>>>(x, psum, s, ss);
  k_select<<<NB, 256, 0, stream>>>(psum, s, ss, sel);
  k_apply<<<NB * NL * NC, 256, 0, stream>>>(x, noise, sel, (float*)d_out);
}